// SAGENet_40518721470745
// MI455X (gfx1250) — compile-verified
//
#include <hip/hip_runtime.h>
#include <math.h>

typedef float v2f __attribute__((ext_vector_type(2)));
typedef float v8f __attribute__((ext_vector_type(8)));

#define N_NODES 100000
#define N_EDGES 1200000
#define F_IN    64
#define F_HID   16
#define F_OUT   40
#define N_TILES (N_NODES / 16)   // 6250, exact

// workspace layout in floats
#define OFF_DEG   0
#define OFF_AGG1  (N_NODES)                     // 100,000
#define OFF_H     (OFF_AGG1 + N_NODES * F_IN)   // 6,500,000
#define OFF_AGG2  (OFF_H + N_NODES * F_HID)     // 8,100,000
#define WS_FLOATS (OFF_AGG2 + N_NODES * F_HID)  // 9,700,000

__global__ void zero_ws_kernel(float* __restrict__ ws, int n4) {
    int i = blockIdx.x * blockDim.x + threadIdx.x;
    if (i < n4) ((float4*)ws)[i] = make_float4(0.f, 0.f, 0.f, 0.f);
}

__global__ void deg_kernel(const int* __restrict__ dst, float* __restrict__ deg) {
    int e = blockIdx.x * blockDim.x + threadIdx.x;
    if (e < N_EDGES) atomicAdd(&deg[dst[e]], 1.0f);
}

// one thread per (edge, 4-feature chunk): 16 chunks of F_IN=64
__global__ void scatter1_kernel(const float* __restrict__ x,
                                const int* __restrict__ src,
                                const int* __restrict__ dst,
                                float* __restrict__ agg1) {
    int t = blockIdx.x * blockDim.x + threadIdx.x;
    if (t >= N_EDGES * 16) return;
    int e = t >> 4;
    int c = t & 15;
    int s = src[e];
    int d = dst[e];
    float4 v = *(const float4*)(x + (size_t)s * F_IN + c * 4);
    float* ap = agg1 + (size_t)d * F_IN + c * 4;
    atomicAdd(ap + 0, v.x);
    atomicAdd(ap + 1, v.y);
    atomicAdd(ap + 2, v.z);
    atomicAdd(ap + 3, v.w);
}

// one thread per (edge, 4-feature chunk): 4 chunks of F_HID=16
__global__ void scatter2_kernel(const float* __restrict__ h,
                                const int* __restrict__ src,
                                const int* __restrict__ dst,
                                float* __restrict__ agg2) {
    int t = blockIdx.x * blockDim.x + threadIdx.x;
    if (t >= N_EDGES * 4) return;
    int e = t >> 2;
    int c = t & 3;
    int s = src[e];
    int d = dst[e];
    float4 v = *(const float4*)(h + (size_t)s * F_HID + c * 4);
    float* ap = agg2 + (size_t)d * F_HID + c * 4;
    atomicAdd(ap + 0, v.x);
    atomicAdd(ap + 1, v.y);
    atomicAdd(ap + 2, v.z);
    atomicAdd(ap + 3, v.w);
}

// One wave per 16-node tile: h = relu(x@W1s + (agg1/deg)@W1n + b1)
__global__ void layer1_wmma_kernel(const float* __restrict__ x,
                                   const float* __restrict__ agg1,
                                   const float* __restrict__ deg,
                                   const float* __restrict__ W1s,
                                   const float* __restrict__ W1n,
                                   const float* __restrict__ b1,
                                   float* __restrict__ h) {
    int wave = threadIdx.x >> 5;
    int lane = threadIdx.x & 31;
    int tile = blockIdx.x * (blockDim.x >> 5) + wave;
    if (tile >= N_TILES) return;
    int nb   = tile * 16;
    int m    = lane & 15;               // A row / B col for this lane
    int koff = (lane >> 4) << 1;        // upper half-wave holds K=2,3 of each step
    int n    = m;

    const float* xrow = x    + (size_t)(nb + m) * F_IN;
    const float* arow = agg1 + (size_t)(nb + m) * F_IN;
    float invd = 1.0f / fmaxf(deg[nb + m], 1.0f);

    v8f acc = {};
#pragma unroll
    for (int k0 = 0; k0 < F_IN; k0 += 4) {
        v2f a, b;
        a[0] = xrow[k0 + koff];
        a[1] = xrow[k0 + koff + 1];
        b[0] = W1s[(k0 + koff) * F_HID + n];
        b[1] = W1s[(k0 + koff + 1) * F_HID + n];
        acc = __builtin_amdgcn_wmma_f32_16x16x4_f32(false, a, false, b,
                                                    (short)0, acc, false, false);
    }
#pragma unroll
    for (int k0 = 0; k0 < F_IN; k0 += 4) {
        v2f a, b;
        a[0] = arow[k0 + koff] * invd;
        a[1] = arow[k0 + koff + 1] * invd;
        b[0] = W1n[(k0 + koff) * F_HID + n];
        b[1] = W1n[(k0 + koff + 1) * F_HID + n];
        acc = __builtin_amdgcn_wmma_f32_16x16x4_f32(false, a, false, b,
                                                    (short)0, acc, false, false);
    }
    float bias  = b1[n];
    int   mbase = (lane >> 4) * 8;
#pragma unroll
    for (int r = 0; r < 8; ++r) {
        float v = fmaxf(acc[r] + bias, 0.0f);
        h[(size_t)(nb + mbase + r) * F_HID + n] = v;
    }
}

// One wave per 16-node tile: out = h@W2s + (agg2/deg)@W2n + b2  (N=40 -> 3 tiles)
__global__ void layer2_wmma_kernel(const float* __restrict__ h,
                                   const float* __restrict__ agg2,
                                   const float* __restrict__ deg,
                                   const float* __restrict__ W2s,
                                   const float* __restrict__ W2n,
                                   const float* __restrict__ b2,
                                   float* __restrict__ out) {
    int wave = threadIdx.x >> 5;
    int lane = threadIdx.x & 31;
    int tile = blockIdx.x * (blockDim.x >> 5) + wave;
    if (tile >= N_TILES) return;
    int nb   = tile * 16;
    int m    = lane & 15;
    int koff = (lane >> 4) << 1;
    int n    = m;

    const float* hrow = h    + (size_t)(nb + m) * F_HID;
    const float* arow = agg2 + (size_t)(nb + m) * F_HID;
    float invd = 1.0f / fmaxf(deg[nb + m], 1.0f);

    // preload A fragments (K=16 -> 4 steps of K=4), reused across 3 N-tiles
    v2f as[4], an[4];
#pragma unroll
    for (int i = 0; i < 4; ++i) {
        int k = 4 * i + koff;
        as[i][0] = hrow[k];
        as[i][1] = hrow[k + 1];
        an[i][0] = arow[k] * invd;
        an[i][1] = arow[k + 1] * invd;
    }
    int mbase = (lane >> 4) * 8;

#pragma unroll
    for (int nt = 0; nt < 3; ++nt) {
        int  col   = nt * 16 + n;
        bool valid = (col < F_OUT);
        v8f  acc   = {};
#pragma unroll
        for (int i = 0; i < 4; ++i) {
            int k = 4 * i + koff;
            v2f b;
            b[0] = valid ? W2s[k * F_OUT + col] : 0.0f;
            b[1] = valid ? W2s[(k + 1) * F_OUT + col] : 0.0f;
            acc = __builtin_amdgcn_wmma_f32_16x16x4_f32(false, as[i], false, b,
                                                        (short)0, acc, false, false);
        }
#pragma unroll
        for (int i = 0; i < 4; ++i) {
            int k = 4 * i + koff;
            v2f b;
            b[0] = valid ? W2n[k * F_OUT + col] : 0.0f;
            b[1] = valid ? W2n[(k + 1) * F_OUT + col] : 0.0f;
            acc = __builtin_amdgcn_wmma_f32_16x16x4_f32(false, an[i], false, b,
                                                        (short)0, acc, false, false);
        }
        if (valid) {
            float bias = b2[col];
#pragma unroll
            for (int r = 0; r < 8; ++r)
                out[(size_t)(nb + mbase + r) * F_OUT + col] = acc[r] + bias;
        }
    }
}

// one wave (32 lanes) per row of 40; in-place log_softmax on d_out
__global__ void logsoftmax_kernel(float* __restrict__ out) {
    int wave = threadIdx.x >> 5;
    int lane = threadIdx.x & 31;
    int row  = blockIdx.x * (blockDim.x >> 5) + wave;
    if (row >= N_NODES) return;
    float* p = out + (size_t)row * F_OUT;
    float v0 = p[lane];
    bool  has2 = (lane < F_OUT - 32);   // 8 extra elements
    float v1 = has2 ? p[lane + 32] : -INFINITY;

    float mx = fmaxf(v0, v1);
#pragma unroll
    for (int s = 16; s > 0; s >>= 1) mx = fmaxf(mx, __shfl_xor(mx, s, 32));

    float sum = expf(v0 - mx) + (has2 ? expf(v1 - mx) : 0.0f);
#pragma unroll
    for (int s = 16; s > 0; s >>= 1) sum += __shfl_xor(sum, s, 32);

    float lse = mx + logf(sum);
    p[lane] = v0 - lse;
    if (has2) p[lane + 32] = v1 - lse;
}

extern "C" void kernel_launch(void* const* d_in, const int* in_sizes, int n_in,
                              void* d_out, int out_size, void* d_ws, size_t ws_size,
                              hipStream_t stream) {
    const float* x   = (const float*)d_in[0];
    const int*   ei  = (const int*)d_in[1];
    const float* W1s = (const float*)d_in[2];
    const float* W1n = (const float*)d_in[3];
    const float* b1  = (const float*)d_in[4];
    const float* W2s = (const float*)d_in[5];
    const float* W2n = (const float*)d_in[6];
    const float* b2  = (const float*)d_in[7];
    const int* src = ei;
    const int* dst = ei + N_EDGES;

    float* out  = (float*)d_out;
    float* ws   = (float*)d_ws;
    float* deg  = ws + OFF_DEG;
    float* agg1 = ws + OFF_AGG1;
    float* h    = ws + OFF_H;
    float* agg2 = ws + OFF_AGG2;

    const int n4 = WS_FLOATS / 4;
    zero_ws_kernel<<<(n4 + 255) / 256, 256, 0, stream>>>(ws, n4);

    deg_kernel<<<(N_EDGES + 255) / 256, 256, 0, stream>>>(dst, deg);

    scatter1_kernel<<<(N_EDGES * 16 + 255) / 256, 256, 0, stream>>>(x, src, dst, agg1);

    layer1_wmma_kernel<<<(N_TILES + 7) / 8, 256, 0, stream>>>(x, agg1, deg,
                                                              W1s, W1n, b1, h);

    scatter2_kernel<<<(N_EDGES * 4 + 255) / 256, 256, 0, stream>>>(h, src, dst, agg2);

    layer2_wmma_kernel<<<(N_TILES + 7) / 8, 256, 0, stream>>>(h, agg2, deg,
                                                              W2s, W2n, b2, out);

    logsoftmax_kernel<<<(N_NODES + 7) / 8, 256, 0, stream>>>(out);
}